// ScaledDotProductAttention_55181739819366
// MI455X (gfx1250) — compile-verified
//
#include <hip/hip_runtime.h>
#include <cstdint>
#include <cstddef>

#define S_LEN   2048
#define DK      64
#define WIN     128
#define BHEADS  16
#define FLT_BIG 3.402823466e38f

typedef float v8f __attribute__((ext_vector_type(8)));
typedef float v2f __attribute__((ext_vector_type(2)));

// ---------------------------------------------------------------------------
// Zero-fill the dense attention output (out-of-band entries must be 0).
// Vectorized float4 stores -> global_store_b128, pure-BW kernel.
// ---------------------------------------------------------------------------
__global__ void zero_attn_kernel(float4* __restrict__ p, size_t n4) {
    size_t i      = (size_t)blockIdx.x * blockDim.x + threadIdx.x;
    size_t stride = (size_t)gridDim.x * blockDim.x;
    float4 z = make_float4(0.f, 0.f, 0.f, 0.f);
    for (; i < n4; i += stride) p[i] = z;
}

// ---------------------------------------------------------------------------
// Compute a 16x16 score tile  S = Q_tile * K_tile^T  with 16 chained
// V_WMMA_F32_16X16X4_F32 steps over D=64.
//   A (16x4 f32): lane m holds Q[q0+m][4s+2h .. +1]  (h = lane>>4)
//   B (4x16 f32): lane n holds K[k0+n][4s+2h .. +1]  (K^T column = K row)
// ---------------------------------------------------------------------------
__device__ __forceinline__ v8f qk_tile(const v2f* qa, const float* __restrict__ Kbh,
                                       int k0, int n, int h) {
    v8f c = {};
#pragma unroll
    for (int s = 0; s < 16; ++s) {
        v2f b = *reinterpret_cast<const v2f*>(Kbh + (size_t)(k0 + n) * DK + 4 * s + 2 * h);
        c = __builtin_amdgcn_wmma_f32_16x16x4_f32(false, qa[s], false, b,
                                                  (short)0, c, false, false);
    }
    return c;
}

__global__ void __launch_bounds__(32)
banded_attn_kernel(const float* __restrict__ Q, const float* __restrict__ K,
                   const float* __restrict__ V, float* __restrict__ ctx,
                   float* __restrict__ attn) {
    // Padded LDS tile: 16 rows x 20 floats (80B row stride -> 16B aligned rows,
    // conflict-spread banks). Used for C-layout -> A-layout transpose of P and
    // as staging for vectorized attn stores.
    __shared__ float lds_p[16 * 20];

    const int lane = threadIdx.x;   // 0..31 (wave32)
    const int n    = lane & 15;     // column index (B/C layout) / row (A layout)
    const int h    = lane >> 4;     // half select
    const int q0   = blockIdx.x * 16;
    const int bh   = blockIdx.y;

    const float* Qbh = Q + (size_t)bh * S_LEN * DK;
    const float* Kbh = K + (size_t)bh * S_LEN * DK;
    const float* Vbh = V + (size_t)bh * S_LEN * DK;

    // Preload Q tile in A-operand layout: qa[s] = Q[q0+n][4s+2h .. +1]
    v2f qa[16];
#pragma unroll
    for (int s = 0; s < 16; ++s)
        qa[s] = *reinterpret_cast<const v2f*>(Qbh + (size_t)(q0 + n) * DK + 4 * s + 2 * h);

    // In-band K tile range for queries [q0, q0+15]
    int lo = q0 - (WIN - 1); if (lo < 0) lo = 0;
    int hi = q0 + 15 + (WIN - 1); if (hi > S_LEN - 1) hi = S_LEN - 1;
    const int t_lo = lo >> 4, t_hi = hi >> 4;

    const float scale = 0.125f;  // 1/sqrt(64)

    // Streaming softmax stats; element r <-> query row q0 + r + 8h
    float m_run[8], l_run[8];
#pragma unroll
    for (int r = 0; r < 8; ++r) { m_run[r] = -FLT_BIG; l_run[r] = 0.f; }

    // ---------------- Pass 1: row max / sum of exp over the band -----------
    for (int t = t_lo; t <= t_hi; ++t) {
        const int k0 = t * 16;
        if (t < t_hi)  // prefetch next K tile rows (speculative, b8)
            __builtin_prefetch(Kbh + (size_t)(k0 + 16 + n) * DK, 0, 1);

        v8f c = qk_tile(qa, Kbh, k0, n, h);
#pragma unroll
        for (int r = 0; r < 8; ++r) {
            const int row = q0 + r + 8 * h;
            const int col = k0 + n;
            const bool inb = (row - col < WIN) && (col - row < WIN);
            const float sv = c[r] * scale;

            float sm = inb ? sv : -FLT_BIG;
#pragma unroll
            for (int off = 1; off < 16; off <<= 1)
                sm = fmaxf(sm, __shfl_xor(sm, off, 16));
            float e = inb ? __expf(sv - sm) : 0.f;
#pragma unroll
            for (int off = 1; off < 16; off <<= 1)
                e += __shfl_xor(e, off, 16);

            const float mnew = fmaxf(m_run[r], sm);
            l_run[r] = l_run[r] * __expf(m_run[r] - mnew) + e * __expf(sm - mnew);
            m_run[r] = mnew;
        }
    }

    float inv_l[8];
#pragma unroll
    for (int r = 0; r < 8; ++r) inv_l[r] = 1.0f / l_run[r];

    v8f acc[4];
    {
        v8f z = {};
#pragma unroll
        for (int dt = 0; dt < 4; ++dt) acc[dt] = z;
    }

    // ---------------- Pass 2: write attn tiles + P·V via WMMA --------------
    float* attn_bh = attn + (size_t)bh * S_LEN * S_LEN;
    for (int t = t_lo; t <= t_hi; ++t) {
        const int k0 = t * 16;
        if (t < t_hi)
            __builtin_prefetch(Vbh + (size_t)(k0 + 16 + n) * DK, 0, 1);

        v8f c = qk_tile(qa, Kbh, k0, n, h);
#pragma unroll
        for (int r = 0; r < 8; ++r) {
            const int row = q0 + r + 8 * h;
            const int col = k0 + n;
            const bool inb = (row - col < WIN) && (col - row < WIN);
            const float sv = c[r] * scale;
            const float p  = inb ? __expf(sv - m_run[r]) * inv_l[r] : 0.f;
            lds_p[(r + 8 * h) * 20 + n] = p;   // C-layout scatter into LDS
        }

        // Coalesced attn store from LDS (same wave; DS ops are in-order).
        {
            const int rr = n;        // row 0..15
            const int cb = 8 * h;    // col base 0 or 8
            const float4 w0 = *reinterpret_cast<const float4*>(&lds_p[rr * 20 + cb]);
            const float4 w1 = *reinterpret_cast<const float4*>(&lds_p[rr * 20 + cb + 4]);
            float* dst = attn_bh + (size_t)(q0 + rr) * S_LEN + k0 + cb;
            *reinterpret_cast<float4*>(dst)     = w0;
            *reinterpret_cast<float4*>(dst + 4) = w1;
        }

        // P(16x16) x V(16x64): 4 K-steps x 4 d-tiles of f32 WMMA.
#pragma unroll
        for (int s = 0; s < 4; ++s) {
            // A operand from LDS (transposed P): lane n = row, k = 4s+2h..+1
            v2f a = *reinterpret_cast<const v2f*>(&lds_p[n * 20 + 4 * s + 2 * h]);
            const float* vp = Vbh + (size_t)(k0 + 4 * s + 2 * h) * DK + n;
#pragma unroll
            for (int dt = 0; dt < 4; ++dt) {
                v2f b;
                b.x = vp[dt * 16];        // B[2h]  [n] = V[k0+4s+2h  ][16dt+n]
                b.y = vp[dt * 16 + DK];   // B[2h+1][n] = V[k0+4s+2h+1][16dt+n]
                acc[dt] = __builtin_amdgcn_wmma_f32_16x16x4_f32(false, a, false, b,
                                                                (short)0, acc[dt],
                                                                false, false);
            }
        }
    }

    // ---------------- Context store (C layout -> [B,H,S,D]) ----------------
    float* ctx_bh = ctx + (size_t)bh * S_LEN * DK;
#pragma unroll
    for (int dt = 0; dt < 4; ++dt)
#pragma unroll
        for (int r = 0; r < 8; ++r)
            ctx_bh[(size_t)(q0 + r + 8 * h) * DK + dt * 16 + n] = acc[dt][r];
}

// ---------------------------------------------------------------------------
extern "C" void kernel_launch(void* const* d_in, const int* in_sizes, int n_in,
                              void* d_out, int out_size, void* d_ws, size_t ws_size,
                              hipStream_t stream) {
    (void)in_sizes; (void)n_in; (void)out_size; (void)d_ws; (void)ws_size;

    const float* Q = (const float*)d_in[0];
    const float* K = (const float*)d_in[1];
    const float* V = (const float*)d_in[2];

    float* ctx  = (float*)d_out;                               // [B,H,S,D]
    float* attn = ctx + (size_t)BHEADS * S_LEN * DK;           // [B,H,S,S]

    // 1) Zero the dense attention output (band kernel only writes in-band tiles).
    const size_t n4 = (size_t)BHEADS * S_LEN * S_LEN / 4;
    zero_attn_kernel<<<2048, 256, 0, stream>>>(reinterpret_cast<float4*>(attn), n4);

    // 2) Banded flash-style attention: one wave32 per 16-query tile.
    dim3 grid(S_LEN / 16, BHEADS);
    banded_attn_kernel<<<grid, 32, 0, stream>>>(Q, K, V, ctx, attn);
}